// ReductionCollision_74577812128040
// MI455X (gfx1250) — compile-verified
//
#include <hip/hip_runtime.h>
#include <stdint.h>

typedef __attribute__((ext_vector_type(2))) float v2f;
typedef __attribute__((ext_vector_type(8))) float v8f;

#define Ndim 256      // full velocity basis
#define Mdim 64       // reduced basis
#define TB   128      // batch rows per workgroup
#define ROWS_PER_WAVE 32
#define LDSS 66       // padded LDS row stride (floats): even (8B align) + bank rotation

// One 16B async global->LDS copy. IOFFSET is added to BOTH the LDS address and
// the global address (CDNA5 ISA pseudocode), so a single LDS-offset
// VGPR + single 64-bit global address VGPR pair cover all chunks.
#define ASYNC_LD16(ldsoff, gptr, IMM)                                          \
    asm volatile("global_load_async_to_lds_b128 %0, %1, off offset:" #IMM     \
                 :: "v"(ldsoff), "v"(gptr) : "memory")

__device__ __forceinline__ void stage_k_slice(const float* __restrict__ gsrc,
                                              unsigned lds_off) {
    // 128 threads x 8 chunks x 16B = 16 KB slice; tracked by ASYNCcnt.
    ASYNC_LD16(lds_off, gsrc, 0);
    ASYNC_LD16(lds_off, gsrc, 2048);
    ASYNC_LD16(lds_off, gsrc, 4096);
    ASYNC_LD16(lds_off, gsrc, 6144);
    ASYNC_LD16(lds_off, gsrc, 8192);
    ASYNC_LD16(lds_off, gsrc, 10240);
    ASYNC_LD16(lds_off, gsrc, 12288);
    ASYNC_LD16(lds_off, gsrc, 14336);
}

__device__ __forceinline__ v8f vzero8() {
    v8f v;
#pragma unroll
    for (int i = 0; i < 8; ++i) v[i] = 0.0f;
    return v;
}

__device__ __forceinline__ v8f wmma_f32(v2f a, v2f b, v8f c) {
    // V_WMMA_F32_16X16X4_F32 : D = A(16x4) x B(4x16) + C(16x16), full fp32
    return __builtin_amdgcn_wmma_f32_16x16x4_f32(
        /*neg_a=*/false, a, /*neg_b=*/false, b,
        /*c_mod=*/(short)0, c, /*reuse_a=*/false, /*reuse_b=*/false);
}

__global__ __launch_bounds__(128) void collision_fused_kernel(
    const float* __restrict__ f,   // (B, 256)
    const float* __restrict__ Fm,  // (256, 64)
    const float* __restrict__ Gm,  // (256, 64)
    const float* __restrict__ Km,  // (64, 64, 64)
    float* __restrict__ out)       // (B, 256)
{
    __shared__ float lds_ff[TB * LDSS];        // ff tile, padded
    __shared__ float lds_q [TB * LDSS];        // Q  tile, padded
    __shared__ float kbuf[2][Mdim * Mdim];     // double-buffered K[i] slices (2 x 16KB)

    const int tid  = threadIdx.x;
    const int wave = tid >> 5;
    const int lane = tid & 31;
    const int lrow = lane & 15;   // row/col-within-tile index
    const int lgrp = lane >> 4;   // which K-slot pair this lane holds

    const int wg_row0   = blockIdx.x * TB;
    const int wave_row0 = wave * ROWS_PER_WAVE;   // LDS-relative base row

    // LDS byte offsets of the two K staging buffers (flat addr low 32 bits ==
    // workgroup-relative LDS offset), plus this thread's 16B lane within a slice.
    const unsigned kb_off0 = (unsigned)(uintptr_t)(&kbuf[0][0]) + (unsigned)(tid * 16);
    const unsigned kb_off1 = (unsigned)(uintptr_t)(&kbuf[1][0]) + (unsigned)(tid * 16);

    // Kick off the async stage of K[0] now; it completes under all of phase 1.
    stage_k_slice(Km + tid * 4, kb_off0);

    // ---------------- Phase 1: ff = f @ F ----------------
    v8f acc1[2][4];
#pragma unroll
    for (int rt = 0; rt < 2; ++rt)
#pragma unroll
        for (int jt = 0; jt < 4; ++jt) acc1[rt][jt] = vzero8();

#pragma unroll 1
    for (int kk = 0; kk < Ndim; kk += 4) {
        v2f afrag[2];
#pragma unroll
        for (int rt = 0; rt < 2; ++rt) {
            const int gr = wg_row0 + wave_row0 + rt * 16 + lrow;
            afrag[rt] = *(const v2f*)(f + (size_t)gr * Ndim + kk + 2 * lgrp);
        }
#pragma unroll
        for (int jt = 0; jt < 4; ++jt) {
            const int col = jt * 16 + lrow;
            v2f bfrag;
            bfrag.x = Fm[(kk + 2 * lgrp    ) * Mdim + col];
            bfrag.y = Fm[(kk + 2 * lgrp + 1) * Mdim + col];
#pragma unroll
            for (int rt = 0; rt < 2; ++rt)
                acc1[rt][jt] = wmma_f32(afrag[rt], bfrag, acc1[rt][jt]);
        }
    }
    // Scatter ff fragments to LDS (C-layout: VGPR r <-> rows {r, r+8})
#pragma unroll
    for (int rt = 0; rt < 2; ++rt)
#pragma unroll
        for (int jt = 0; jt < 4; ++jt)
#pragma unroll
            for (int r = 0; r < 8; ++r) {
                const int row = wave_row0 + rt * 16 + r + 8 * lgrp;
                const int col = jt * 16 + lrow;
                lds_ff[row * LDSS + col] = acc1[rt][jt][r];
            }
    __syncthreads();

    // ---------------- Phase 2: Q_bk = sum_ij ff_bi ff_bj K_ijk ----------------
    // GEMM with on-the-fly A: A[b][i*64+j] = ff_bi * ff_bj; B = K[i] slice
    // streamed into LDS once per workgroup via double-buffered async copies.
    v8f accQ[2][4];
#pragma unroll
    for (int rt = 0; rt < 2; ++rt)
#pragma unroll
        for (int kt = 0; kt < 4; ++kt) accQ[rt][kt] = vzero8();

#pragma unroll 1
    for (int i = 0; i < Mdim; ++i) {
        // Slice i landed (my copies), then barrier: everyone's copies landed and
        // everyone is done reading the buffer we are about to overwrite.
        asm volatile("s_wait_asynccnt 0x0" ::: "memory");
        __syncthreads();
        if (i + 1 < Mdim)
            stage_k_slice(Km + (size_t)(i + 1) * Mdim * Mdim + tid * 4,
                          ((i + 1) & 1) ? kb_off1 : kb_off0);

        const float* Kb = &kbuf[i & 1][0];

        float ffi[2];
#pragma unroll
        for (int rt = 0; rt < 2; ++rt)
            ffi[rt] = lds_ff[(wave_row0 + rt * 16 + lrow) * LDSS + i];

#pragma unroll 1
        for (int j0 = 0; j0 < Mdim; j0 += 4) {
            v2f afrag[2];
#pragma unroll
            for (int rt = 0; rt < 2; ++rt) {
                v2f t = *(const v2f*)&lds_ff[(wave_row0 + rt * 16 + lrow) * LDSS + j0 + 2 * lgrp];
                afrag[rt] = ffi[rt] * t;   // outer-product A fragment, zero memory traffic
            }
            const float* Kp = Kb + (size_t)j0 * Mdim;
#pragma unroll
            for (int kt = 0; kt < 4; ++kt) {
                const int col = kt * 16 + lrow;
                v2f bfrag;
                bfrag.x = Kp[(2 * lgrp    ) * Mdim + col];   // LDS reads, 2addr-fusable
                bfrag.y = Kp[(2 * lgrp + 1) * Mdim + col];
#pragma unroll
                for (int rt = 0; rt < 2; ++rt)
                    accQ[rt][kt] = wmma_f32(afrag[rt], bfrag, accQ[rt][kt]);
            }
        }
    }
    // Scatter Q to LDS for re-layout into A fragments
#pragma unroll
    for (int rt = 0; rt < 2; ++rt)
#pragma unroll
        for (int kt = 0; kt < 4; ++kt)
#pragma unroll
            for (int r = 0; r < 8; ++r) {
                const int row = wave_row0 + rt * 16 + r + 8 * lgrp;
                const int col = kt * 16 + lrow;
                lds_q[row * LDSS + col] = accQ[rt][kt][r];
            }
    __syncthreads();

    // ---------------- Phase 3: Qr = Q @ G^T ----------------
#pragma unroll 1
    for (int nt = 0; nt < Ndim / 16; ++nt) {
        v8f accO[2];
#pragma unroll
        for (int rt = 0; rt < 2; ++rt) accO[rt] = vzero8();

#pragma unroll 1
        for (int k0 = 0; k0 < Mdim; k0 += 4) {
            // B[k][n] = G[n][k] -> contiguous pair in G's row
            v2f bfrag = *(const v2f*)&Gm[(size_t)(nt * 16 + lrow) * Mdim + k0 + 2 * lgrp];
#pragma unroll
            for (int rt = 0; rt < 2; ++rt) {
                v2f afrag = *(const v2f*)&lds_q[(wave_row0 + rt * 16 + lrow) * LDSS + k0 + 2 * lgrp];
                accO[rt] = wmma_f32(afrag, bfrag, accO[rt]);
            }
        }
#pragma unroll
        for (int rt = 0; rt < 2; ++rt)
#pragma unroll
            for (int r = 0; r < 8; ++r) {
                const int gr = wg_row0 + wave_row0 + rt * 16 + r + 8 * lgrp;
                out[(size_t)gr * Ndim + nt * 16 + lrow] = accO[rt][r];
            }
    }
}

extern "C" void kernel_launch(void* const* d_in, const int* in_sizes, int n_in,
                              void* d_out, int out_size, void* d_ws, size_t ws_size,
                              hipStream_t stream) {
    const float* f  = (const float*)d_in[0];   // (B, 256)
    const float* Fm = (const float*)d_in[1];   // (256, 64)
    const float* Gm = (const float*)d_in[2];   // (256, 64)
    const float* Km = (const float*)d_in[3];   // (64, 64, 64)
    float* out = (float*)d_out;                // (B, 256)

    const int rowsB  = in_sizes[0] / Ndim;     // 65536
    const int blocks = rowsB / TB;             // 512 workgroups, 4 waves each

    collision_fused_kernel<<<blocks, 128, 0, stream>>>(f, Fm, Gm, Km, out);
}